// DetectionLoss_6700148982213
// MI455X (gfx1250) — compile-verified
//
#include <hip/hip_runtime.h>
#include <stdint.h>

// ---------------- constants (DetectionLoss defaults) ----------------
#define BATCH     16
#define NUMNEG    10000
#define NUMHARD   100
#define NEGRATIO  100u

typedef float v2f __attribute__((ext_vector_type(2)));
typedef float v8f __attribute__((ext_vector_type(8)));

// ---------------- workspace layout (uint32 offsets) ----------------
#define OFF_HIST1    0u          /* BATCH*2048 */
#define OFF_HIST2    32768u      /* BATCH*4096 */
#define OFF_POSSUM   98304u      /* BATCH floats */
#define OFF_NPOS     98320u      /* BATCH floats */
#define OFF_TIECNT   98336u      /* BATCH */
#define OFF_SLOTCNT  98352u      /* BATCH */
#define OFF_NEGSEL   98368u      /* BATCH floats */
#define ZERO_COUNT   98384u
#define OFF_KEYS     98384u      /* BATCH*2 */
#define OFF_P1       98416u
#define OFF_R1       98432u
#define OFF_PK       98448u
#define OFF_R2       98464u
#define OFF_K        98480u
#define OFF_DENOM    98496u      /* BATCH floats */
#define OFF_SAMPLED  98512u      /* BATCH*NUMNEG floats */

// ---------------- threefry2x32 (matches jax.random default PRNG) ----
static __device__ __forceinline__ uint2 threefry2x32(uint2 k, uint2 c) {
  uint32_t ks[3] = {k.x, k.y, k.x ^ k.y ^ 0x1BD11BDAu};
  uint32_t x0 = c.x + ks[0];
  uint32_t x1 = c.y + ks[1];
  const int rot[2][4] = {{13, 15, 26, 6}, {17, 29, 16, 24}};
#pragma unroll
  for (int g = 0; g < 5; ++g) {
#pragma unroll
    for (int i = 0; i < 4; ++i) {
      int r = rot[g & 1][i];
      x0 += x1;
      x1 = (x1 << r) | (x1 >> (32 - r));
      x1 ^= x0;
    }
    x0 += ks[(g + 1) % 3];
    x1 += ks[(g + 2) % 3] + (uint32_t)(g + 1);
  }
  return make_uint2(x0, x1);
}

// ---------------- math helpers ----------------
static __device__ __forceinline__ float sigp(float x) {
  float p = 1.f / (1.f + expf(-x));
  return fminf(fmaxf(p, 1e-4f), 1.f - 1e-4f);
}
static __device__ __forceinline__ float bce_logits(float x, float t) {
  return fmaxf(x, 0.f) - x * t + log1pf(expf(-fabsf(x)));
}
static __device__ __forceinline__ float neg_loss(float x, float m) {
  float p = sigp(x);
  float loss = (m == 0.f) ? 0.25f * p * p * bce_logits(x, 0.f) : 0.f;
  if (p > 0.5f) {
    float u = fminf(fmaxf((p - 0.5f) * 5.f, 0.f), 1.f);
    loss *= 1.5f + 0.5f * u;
  }
  return loss;
}

// 64-way wave reduction via V_WMMA_F32_16X16X4_F32: A holds the 64
// accumulators (2 VGPRs x 32 lanes, each value appears exactly once in the
// 16x4 A tile), B = ones => D[i][j] = rowsum_i. Lanes 0-15 hold rows 0-7 in
// D VGPRs 0..7, lanes 16-31 hold rows 8-15; per-lane sum + shfl_xor(16)
// yields the total independent of the exact A striping.
static __device__ __forceinline__ float wave_reduce_wmma(v2f acc) {
  v8f c = {};
  v2f ones = {1.f, 1.f};
  v8f d = __builtin_amdgcn_wmma_f32_16x16x4_f32(
      false, acc, false, ones, (short)0, c, false, false);
  float s = d[0] + d[1] + d[2] + d[3] + d[4] + d[5] + d[6] + d[7];
  s += __shfl_xor(s, 16, 32);
  return s;
}

// ---------------- kernels ----------------
__global__ void k_zero(uint32_t* w, int n) {
  int i = blockIdx.x * blockDim.x + threadIdx.x;
  if (i < n) w[i] = 0u;
}

// jax.random.split(key(42), 16): counts=arange(32) split into halves;
// out[m<16]=r_m.x, out[m>=16]=r_{m-16}.y ; keys[b]=(out[2b],out[2b+1])
__global__ void k_keys(uint32_t* keys) {
  __shared__ uint32_t outA[32];
  int m = threadIdx.x;
  if (m < 16) {
    uint2 r = threefry2x32(make_uint2(0u, 42u), make_uint2((uint32_t)m, (uint32_t)(m + 16)));
    outA[m] = r.x;
    outA[m + 16] = r.y;
  }
  __syncthreads();
  if (m < 16) {
    keys[2 * m]     = outA[2 * m];
    keys[2 * m + 1] = outA[2 * m + 1];
  }
}

// Pass A: dense read of target; sparse pred/mask for positives.
// Produces: pos-loss sum, npos, level-1 (2048-bin) histogram of rnd key23
// over negatives. WMMA wave reduction for the two accumulators.
__global__ void __launch_bounds__(256)
k_passA(const float* __restrict__ pred, const float* __restrict__ tgt,
        const float* __restrict__ msk, const uint32_t* __restrict__ keys,
        uint32_t* __restrict__ hist1, float* __restrict__ posSum,
        float* __restrict__ nposF, int N) {
  __shared__ uint32_t h[2048];
  const int b = blockIdx.y;
  const int half = N >> 1;
  for (int i = threadIdx.x; i < 2048; i += blockDim.x) h[i] = 0u;
  __syncthreads();
  const uint2 key = make_uint2(keys[2 * b], keys[2 * b + 1]);
  const size_t base = (size_t)b * (size_t)N;
  v2f accL = {0.f, 0.f};
  v2f accC = {0.f, 0.f};
  for (int j = blockIdx.x * blockDim.x + threadIdx.x; j < half;
       j += gridDim.x * blockDim.x) {
    uint2 r = threefry2x32(key, make_uint2((uint32_t)j, (uint32_t)(j + half)));
#pragma unroll
    for (int e = 0; e < 2; ++e) {
      size_t idx = base + (size_t)j + (size_t)(e ? half : 0);
      uint32_t bits = e ? r.y : r.x;
      float t = tgt[idx];
      if (t == 1.f) {
        float x = pred[idx];
        float m = msk[idx];
        float p = sigp(x);
        float w = 0.75f * (1.f - p) * (1.f - p);
        float loss = (m == 0.f) ? w * bce_logits(x, 1.f) : 0.f;
        if (p < 0.8f) loss *= 4.f;              // FN upweight (pos only)
        if (e) { accL.y += loss; accC.y += 1.f; }
        else   { accL.x += loss; accC.x += 1.f; }
      } else if (t == 0.f) {
        atomicAdd(&h[(bits >> 9) >> 12], 1u);   // key23 top 11 bits
      }
    }
  }
  __syncthreads();
  for (int i = threadIdx.x; i < 2048; i += blockDim.x)
    if (h[i]) atomicAdd(&hist1[b * 2048 + i], h[i]);
  float tL = wave_reduce_wmma(accL);
  float tC = wave_reduce_wmma(accC);
  if ((threadIdx.x & 31u) == 0u) {
    atomicAdd(&posSum[b], tL);
    atomicAdd(&nposF[b], tC);
  }
}

__global__ void k_pivot1(const uint32_t* hist1, uint32_t* p1, uint32_t* R1) {
  int b = threadIdx.x;
  if (b >= BATCH) return;
  const uint32_t* h = hist1 + b * 2048;
  uint32_t cum = 0, bin = 0, rem = 0xFFFFFFFFu;
  bool found = false;
  for (int i = 2047; i >= 0 && !found; --i) {
    uint32_t c = h[i];
    if (cum + c >= (uint32_t)NUMNEG) { bin = (uint32_t)i; rem = NUMNEG - cum; found = true; }
    else cum += c;
  }
  p1[b] = bin;
  R1[b] = rem;
}

// Pass B: dense read of target only; level-2 (4096-bin) histogram within
// the pivot level-1 bin.
__global__ void __launch_bounds__(256)
k_passB(const float* __restrict__ tgt, const uint32_t* __restrict__ keys,
        const uint32_t* __restrict__ p1arr, uint32_t* __restrict__ hist2, int N) {
  __shared__ uint32_t h[4096];
  const int b = blockIdx.y;
  const int half = N >> 1;
  for (int i = threadIdx.x; i < 4096; i += blockDim.x) h[i] = 0u;
  __syncthreads();
  const uint2 key = make_uint2(keys[2 * b], keys[2 * b + 1]);
  const uint32_t p1 = p1arr[b];
  const size_t base = (size_t)b * (size_t)N;
  for (int j = blockIdx.x * blockDim.x + threadIdx.x; j < half;
       j += gridDim.x * blockDim.x) {
    uint2 r = threefry2x32(key, make_uint2((uint32_t)j, (uint32_t)(j + half)));
#pragma unroll
    for (int e = 0; e < 2; ++e) {
      size_t idx = base + (size_t)j + (size_t)(e ? half : 0);
      uint32_t k23 = (e ? r.y : r.x) >> 9;
      if (tgt[idx] == 0.f && (k23 >> 12) == p1)
        atomicAdd(&h[k23 & 0xFFFu], 1u);
    }
  }
  __syncthreads();
  for (int i = threadIdx.x; i < 4096; i += blockDim.x)
    if (h[i]) atomicAdd(&hist2[b * 4096 + i], h[i]);
}

__global__ void k_pivot2(const uint32_t* hist2, const uint32_t* p1,
                         const uint32_t* R1, const float* nposF,
                         uint32_t* pivotKey, uint32_t* R2, uint32_t* kArr,
                         float* denom) {
  int b = threadIdx.x;
  if (b >= BATCH) return;
  const uint32_t* h = hist2 + b * 4096;
  uint32_t rem = R1[b];
  uint32_t cum = 0, bin = 0, r2 = 0xFFFFFFFFu;
  bool found = false;
  for (int i = 4095; i >= 0 && !found; --i) {
    uint32_t c = h[i];
    if (cum + c >= rem) { bin = (uint32_t)i; r2 = rem - cum; found = true; }
    else cum += c;
  }
  pivotKey[b] = (p1[b] << 12) | bin;   // exact 23-bit pivot key
  R2[b] = r2;                          // #elements to take at exact pivot
  uint32_t np = (uint32_t)(nposF[b] + 0.5f);
  kArr[b] = (np > 0u) ? min(NEGRATIO * np, (uint32_t)NUMNEG) : (uint32_t)NUMHARD;
  denom[b] = (np > 0u) ? fmaxf((float)np, 1.f) : 1.f;
}

// Pass C: dense read of target; sparse pred/mask for selected negatives
// (key23 > pivot, plus R2 ticketed ties). Compacts exactly NUMNEG losses.
__global__ void __launch_bounds__(256)
k_passC(const float* __restrict__ pred, const float* __restrict__ tgt,
        const float* __restrict__ msk, const uint32_t* __restrict__ keys,
        const uint32_t* __restrict__ pivotKey, const uint32_t* __restrict__ R2,
        uint32_t* __restrict__ tieCnt, uint32_t* __restrict__ slotCnt,
        float* __restrict__ sampled, int N) {
  const int b = blockIdx.y;
  const int half = N >> 1;
  const uint2 key = make_uint2(keys[2 * b], keys[2 * b + 1]);
  const uint32_t pk = pivotKey[b];
  const uint32_t r2 = R2[b];
  const size_t base = (size_t)b * (size_t)N;
  for (int j = blockIdx.x * blockDim.x + threadIdx.x; j < half;
       j += gridDim.x * blockDim.x) {
    uint2 r = threefry2x32(key, make_uint2((uint32_t)j, (uint32_t)(j + half)));
#pragma unroll
    for (int e = 0; e < 2; ++e) {
      size_t idx = base + (size_t)j + (size_t)(e ? half : 0);
      uint32_t k23 = (e ? r.y : r.x) >> 9;
      if (tgt[idx] != 0.f) continue;
      if (k23 < pk) continue;
      bool sel = (k23 > pk);
      if (!sel) sel = (atomicAdd(&tieCnt[b], 1u) < r2);
      if (!sel) continue;
      float loss = neg_loss(pred[idx], msk[idx]);
      uint32_t slot = atomicAdd(&slotCnt[b], 1u);
      if (slot < (uint32_t)NUMNEG)
        sampled[(size_t)b * NUMNEG + slot] = loss;
    }
  }
}

// Per-sample: if k >= #sampled, sum everything; else exact byte-radix
// top-k select on the (non-negative => uint-monotone) loss values.
__global__ void __launch_bounds__(1024)
k_select(const float* __restrict__ sampled, const uint32_t* __restrict__ slotCnt,
         const uint32_t* __restrict__ kArr, float* __restrict__ negSel) {
  __shared__ float buf[NUMNEG];
  __shared__ float red[1024];
  __shared__ uint32_t hist[256];
  __shared__ uint32_t sPrefix, sG;
  const int b = blockIdx.x;
  const int tid = threadIdx.x;
  const uint32_t n = min(slotCnt[b], (uint32_t)NUMNEG);
  for (int i = tid; i < NUMNEG; i += 1024)
    buf[i] = (i < (int)n) ? sampled[(size_t)b * NUMNEG + i] : 0.f;
  if (tid == 0) { sPrefix = 0u; sG = 0u; }
  __syncthreads();
  const uint32_t k = kArr[b];
  float sum = 0.f;
  if (k >= n) {
    for (int i = tid; i < (int)n; i += 1024) sum += buf[i];
    red[tid] = sum;
    __syncthreads();
    for (int s = 512; s > 0; s >>= 1) {
      if (tid < s) red[tid] += red[tid + s];
      __syncthreads();
    }
    if (tid == 0) negSel[b] = red[0];
  } else {
    uint32_t mask = 0u;
    for (int shift = 24; shift >= 0; shift -= 8) {
      for (int i = tid; i < 256; i += 1024) hist[i] = 0u;
      __syncthreads();
      uint32_t pfx = sPrefix;
      for (int i = tid; i < (int)n; i += 1024) {
        uint32_t bits = __float_as_uint(buf[i]);
        if ((bits & mask) == pfx) atomicAdd(&hist[(bits >> shift) & 0xFFu], 1u);
      }
      __syncthreads();
      if (tid == 0) {
        uint32_t need = k - sG;
        uint32_t cum = 0;
        int sel = 0;
        for (int v = 255; v >= 0; --v) {
          uint32_t c = hist[v];
          if (cum + c >= need) { sel = v; break; }
          cum += c;
        }
        sG += cum;
        sPrefix |= ((uint32_t)sel) << shift;
      }
      __syncthreads();
      mask |= (0xFFu << shift);
    }
    const uint32_t T = sPrefix;   // k-th largest value bits
    const uint32_t G = sG;        // count strictly greater than T
    for (int i = tid; i < (int)n; i += 1024)
      if (__float_as_uint(buf[i]) > T) sum += buf[i];
    red[tid] = sum;
    __syncthreads();
    for (int s = 512; s > 0; s >>= 1) {
      if (tid < s) red[tid] += red[tid + s];
      __syncthreads();
    }
    // ties have identical value, so (k-G)*T is exact regardless of choice
    if (tid == 0) negSel[b] = red[0] + (float)(k - G) * __uint_as_float(T);
  }
}

__global__ void k_finalize(const float* posSum, const float* negSel,
                           const float* denom, float* out) {
  int t = threadIdx.x;
  float pv = 0.f, nv = 0.f;
  if (t < BATCH) {
    pv = posSum[t] / denom[t];
    nv = negSel[t] / denom[t];
  }
  for (int o = 16; o > 0; o >>= 1) {
    pv += __shfl_down(pv, o, 32);
    nv += __shfl_down(nv, o, 32);
  }
  if (t == 0) {
    out[0] = pv / (float)BATCH;
    out[1] = nv / (float)BATCH;
  }
}

// ---------------- launcher ----------------
extern "C" void kernel_launch(void* const* d_in, const int* in_sizes, int n_in,
                              void* d_out, int out_size, void* d_ws, size_t ws_size,
                              hipStream_t stream) {
  const float* pred = (const float*)d_in[0];
  const float* tgt  = (const float*)d_in[1];
  const float* msk  = (const float*)d_in[2];
  float* out = (float*)d_out;
  const int N = in_sizes[0] / BATCH;   // 884736

  uint32_t* W = (uint32_t*)d_ws;
  uint32_t* hist1   = W + OFF_HIST1;
  uint32_t* hist2   = W + OFF_HIST2;
  float*    posSum  = (float*)(W + OFF_POSSUM);
  float*    nposF   = (float*)(W + OFF_NPOS);
  uint32_t* tieCnt  = W + OFF_TIECNT;
  uint32_t* slotCnt = W + OFF_SLOTCNT;
  float*    negSel  = (float*)(W + OFF_NEGSEL);
  uint32_t* keys    = W + OFF_KEYS;
  uint32_t* p1      = W + OFF_P1;
  uint32_t* R1      = W + OFF_R1;
  uint32_t* pk      = W + OFF_PK;
  uint32_t* R2      = W + OFF_R2;
  uint32_t* kArr    = W + OFF_K;
  float*    denom   = (float*)(W + OFF_DENOM);
  float*    sampled = (float*)(W + OFF_SAMPLED);

  k_zero<<<(ZERO_COUNT + 255) / 256, 256, 0, stream>>>(W, (int)ZERO_COUNT);
  k_keys<<<1, 32, 0, stream>>>(keys);
  k_passA<<<dim3(256, BATCH), 256, 0, stream>>>(pred, tgt, msk, keys, hist1,
                                                posSum, nposF, N);
  k_pivot1<<<1, 32, 0, stream>>>(hist1, p1, R1);
  k_passB<<<dim3(128, BATCH), 256, 0, stream>>>(tgt, keys, p1, hist2, N);
  k_pivot2<<<1, 32, 0, stream>>>(hist2, p1, R1, nposF, pk, R2, kArr, denom);
  k_passC<<<dim3(256, BATCH), 256, 0, stream>>>(pred, tgt, msk, keys, pk, R2,
                                                tieCnt, slotCnt, sampled, N);
  k_select<<<BATCH, 1024, 0, stream>>>(sampled, slotCnt, kArr, negSel);
  k_finalize<<<1, 32, 0, stream>>>(posSum, negSel, denom, out);
}